// LightGCN_11055245820449
// MI455X (gfx1250) — compile-verified
//
#include <hip/hip_runtime.h>
#include <hip/hip_bf16.h>

#define N_USERS   100000
#define N_ITEMS   50000
#define N_NODES   150000
#define DIM       128
#define N_EDGES   6400000
#define N_LAYERS  3

typedef __attribute__((ext_vector_type(2))) float v2f;
typedef __attribute__((ext_vector_type(8))) float v8f;

// ---------------------------------------------------------------------------
// init: cur = concat(user_w, item_w); acc = same. float4 per thread.
// total float4 = 150000*128/4 = 4,800,000 -> 18750 blocks x 256 exactly.
// ---------------------------------------------------------------------------
__global__ void lgcn_init_kernel(const float* __restrict__ user_w,
                                 const float* __restrict__ item_w,
                                 float* __restrict__ cur,
                                 float* __restrict__ acc) {
    size_t i = (size_t)blockIdx.x * blockDim.x + threadIdx.x;   // float4 idx
    const size_t user_f4 = (size_t)N_USERS * DIM / 4;
    float4 v;
    if (i < user_f4) v = ((const float4*)user_w)[i];
    else             v = ((const float4*)item_w)[i - user_f4];
    ((float4*)cur)[i] = v;
    ((float4*)acc)[i] = v;
}

__global__ void lgcn_zero_kernel(float* __restrict__ p) {
    size_t i = (size_t)blockIdx.x * blockDim.x + threadIdx.x;
    float4 z; z.x = 0.f; z.y = 0.f; z.z = 0.f; z.w = 0.f;
    ((float4*)p)[i] = z;
}

// ---------------------------------------------------------------------------
// SpMM scatter: one wave32 per edge (grid-stride). Each lane handles 4 dims
// (float4 gather, 4x global_atomic_add_f32 scatter). Both tables are
// L2-resident (76.8 MB each vs 192 MB L2), so atomics resolve in L2.
// ---------------------------------------------------------------------------
__global__ void lgcn_scatter_kernel(const int*   __restrict__ esrc,
                                    const int*   __restrict__ edst,
                                    const float* __restrict__ evals,
                                    const float* __restrict__ cur,
                                    float*       __restrict__ nxt) {
    const int lane    = threadIdx.x & 31;
    const int wave    = ((int)blockIdx.x * blockDim.x + threadIdx.x) >> 5;
    const int n_waves = ((int)gridDim.x * blockDim.x) >> 5;
    for (int e = wave; e < N_EDGES; e += n_waves) {
        const int   s = esrc[e];
        const int   d = edst[e];
        const float w = evals[e];
        const float4 x = ((const float4*)(cur + (size_t)s * DIM))[lane];
        float* outp = nxt + (size_t)d * DIM + lane * 4;
        __hip_atomic_fetch_add(outp + 0, x.x * w, __ATOMIC_RELAXED, __HIP_MEMORY_SCOPE_AGENT);
        __hip_atomic_fetch_add(outp + 1, x.y * w, __ATOMIC_RELAXED, __HIP_MEMORY_SCOPE_AGENT);
        __hip_atomic_fetch_add(outp + 2, x.z * w, __ATOMIC_RELAXED, __HIP_MEMORY_SCOPE_AGENT);
        __hip_atomic_fetch_add(outp + 3, x.w * w, __ATOMIC_RELAXED, __HIP_MEMORY_SCOPE_AGENT);
    }
}

// ---------------------------------------------------------------------------
// Dst_tile = bscale * X_tile + C_tile via 4 chained V_WMMA_F32_16X16X4_F32.
// X(16x16) = sum_j A_j * B_j, A_j = X[:,4j:4j+4], B_j[k][n] = bscale*(n==4j+k)
// (exact: each output element is one input element times 1.0 / 0.25).
// One wave per 16x16 tile. 9375 row-tiles x 8 col-tiles = 75000 tiles,
// 8 waves/block -> 9375 blocks exactly (EXEC all-1s, as WMMA requires).
//
// VGPR layouts per CDNA5 ISA 7.12.2 (f32):
//   A 16x4 : lane l -> M=l%16, vgpr v -> K = v + 2*(l/16)
//   B 4x16 : lane l -> N=l%16, vgpr v -> K = v + 2*(l/16)
//   C/D    : lane l -> N=l%16, vgpr r -> M = r + 8*(l/16)
// ---------------------------------------------------------------------------
__global__ void lgcn_wmma_tile_kernel(const float* __restrict__ X,
                                      const float* __restrict__ Cin,   // null => C=0
                                      float*       __restrict__ Dst,
                                      float bscale) {
    const int lane = threadIdx.x & 31;
    const int tile = (int)blockIdx.x * (blockDim.x >> 5) + (threadIdx.x >> 5);
    const int tn   = tile & 7;          // DIM/16 = 8 column tiles
    const int tm   = tile >> 3;         // 0 .. 9374
    const int half = lane >> 4;         // 0 / 1
    const int lmod = lane & 15;
    const size_t row0 = (size_t)tm * 16;
    const size_t col0 = (size_t)tn * 16;

    v8f c = {};
    if (Cin != nullptr) {               // kernel-uniform branch: EXEC stays full
#pragma unroll
        for (int r = 0; r < 8; ++r) {
            const int m = r + 8 * half;
            c[r] = Cin[(row0 + m) * DIM + col0 + lmod];
        }
    }

#pragma unroll
    for (int j = 0; j < 4; ++j) {
        // A_j: two consecutive f32 per lane -> global_load_b64
        const float2 av = *(const float2*)&X[(row0 + lmod) * DIM + col0 + 4 * j + 2 * half];
        v2f a; a.x = av.x; a.y = av.y;
        // B_j: scaled column selector, built in registers
        v2f b;
        b.x = (lmod == 4 * j + 0 + 2 * half) ? bscale : 0.0f;
        b.y = (lmod == 4 * j + 1 + 2 * half) ? bscale : 0.0f;
        c = __builtin_amdgcn_wmma_f32_16x16x4_f32(
                /*neg_a=*/false, a, /*neg_b=*/false, b,
                /*c_mod=*/(short)0, c, /*reuse_a=*/false, /*reuse_b=*/false);
    }

#pragma unroll
    for (int r = 0; r < 8; ++r) {
        const int m = r + 8 * half;
        Dst[(row0 + m) * DIM + col0 + lmod] = c[r];
    }
}

// ---------------------------------------------------------------------------
extern "C" void kernel_launch(void* const* d_in, const int* in_sizes, int n_in,
                              void* d_out, int out_size, void* d_ws, size_t ws_size,
                              hipStream_t stream) {
    const int*   edge_src  = (const int*)  d_in[0];
    const int*   edge_dst  = (const int*)  d_in[1];
    const float* edge_vals = (const float*)d_in[2];
    const float* user_w    = (const float*)d_in[3];
    const float* item_w    = (const float*)d_in[4];

    const size_t nelem = (size_t)N_NODES * DIM;          // 19.2M floats
    float* cur = (float*)d_ws;                           // 76.8 MB
    float* nxt = cur + nelem;                            // 76.8 MB
    float* acc = (float*)d_out;                          // acc lives in d_out

    const int ew_blocks   = 18750;                       // nelem/4/256 exactly
    const int tile_blocks = 9375;                        // 75000 tiles / 8 waves

    lgcn_init_kernel<<<ew_blocks, 256, 0, stream>>>(user_w, item_w, cur, acc);

    for (int layer = 0; layer < N_LAYERS; ++layer) {
        lgcn_zero_kernel<<<ew_blocks, 256, 0, stream>>>(nxt);
        lgcn_scatter_kernel<<<2048, 256, 0, stream>>>(edge_src, edge_dst, edge_vals,
                                                      cur, nxt);
        // acc += nxt  (exact, via WMMA selector product)
        lgcn_wmma_tile_kernel<<<tile_blocks, 256, 0, stream>>>(nxt, acc, acc, 1.0f);
        float* t = cur; cur = nxt; nxt = t;
    }

    // d_out = acc * 0.25 in place (C = 0, B = 0.25*selector)
    lgcn_wmma_tile_kernel<<<tile_blocks, 256, 0, stream>>>(acc, nullptr, acc, 0.25f);
}